// NormalizeFramesSpeed_46832323396062
// MI455X (gfx1250) — compile-verified
//
#include <hip/hip_runtime.h>

typedef __attribute__((ext_vector_type(2))) float v2f;
typedef __attribute__((ext_vector_type(8))) float v8f;

#define FRAMES      8192
#define PEOPLE      2
#define KPTS        543
#define DIMS        3
#define MAXF        512
#define NPAIRS      (FRAMES - 1)            // 8191 movement values
#define EPF         300                     // 2 people * 75 sel keypoints * 2 dims
#define FRAME_ELEMS (PEOPLE * KPTS * DIMS)  // 3258 floats per frame

__device__ __forceinline__ float nz(float v) {
    return __builtin_isnan(v) ? 0.0f : v;   // jnp.nan_to_num(nan=0.0)
}

// Map compacted movement-element index e in [0,300) to a flat offset within a frame.
// e -> person p, selected-keypoint r (0..74), dim d (0..1); keypoint = r<32 ? r : 500+(r-32)
__device__ __forceinline__ int elem_off(int frame, int e) {
    int p   = e / 150;
    int rem = e - p * 150;
    int r   = rem >> 1;
    int d   = rem & 1;
    int kp  = (r < 32) ? r : (468 + r);
    return ((frame * PEOPLE + p) * KPTS + kp) * DIMS + d;
}

// ---------------------------------------------------------------------------
// Kernel 1: movements[f] = sum_e |t[f+1,e] - t[f,e]| via V_WMMA_F32_16X16X4_F32
// One wave owns 16 frame pairs (rows M). Each WMMA step consumes K=4 element
// slots; lane half (lane>>4) supplies K=0,1 or K=2,3. B = ones => D = row sums.
// 75 steps cover all 300 elements; f32 exact.
// ---------------------------------------------------------------------------
__global__ __launch_bounds__(128) void movements_kernel(const float* __restrict__ in,
                                                        float* __restrict__ mov) {
    const int lane     = threadIdx.x & 31;
    const int wave     = threadIdx.x >> 5;
    const int pairBase = (blockIdx.x * 4 + wave) * 16;
    const int m        = lane & 15;
    const int half     = lane >> 4;
    const int pair     = pairBase + m;
    const bool ok      = pair < NPAIRS;
    const int  f0      = ok ? pair : 0;     // clamp address, mask value

    v8f c = {};
    v2f bones; bones.x = 1.0f; bones.y = 1.0f;

    for (int i = 0; i < EPF / 4; ++i) {
        const int e0 = i * 4 + half * 2;
        float a0c = nz(in[elem_off(f0,     e0)]);
        float a0n = nz(in[elem_off(f0 + 1, e0)]);
        float a1c = nz(in[elem_off(f0,     e0 + 1)]);
        float a1n = nz(in[elem_off(f0 + 1, e0 + 1)]);
        v2f a;
        a.x = ok ? __builtin_fabsf(a0n - a0c) : 0.0f;   // arithmetic mask keeps EXEC all-1s
        a.y = ok ? __builtin_fabsf(a1n - a1c) : 0.0f;
        // D = A(16x4) * ones(4x16) + C  -> every column of D holds the row sums
        c = __builtin_amdgcn_wmma_f32_16x16x4_f32(false, a, false, bones,
                                                  (short)0, c, false, false);
    }

    // D layout: VGPR r, lanes 0-15 -> M=r ; lanes 16-31 -> M=r+8 (all N identical)
    if (lane == 0) {
        for (int r = 0; r < 8; ++r) {
            int p = pairBase + r;
            if (p < NPAIRS) mov[p] = c[r];
        }
    } else if (lane == 16) {
        for (int r = 0; r < 8; ++r) {
            int p = pairBase + 8 + r;
            if (p < NPAIRS) mov[p] = c[r];
        }
    }
}

// ---------------------------------------------------------------------------
// Kernel 2: single-block inclusive scan -> cum[0..8191], cum[0] = 0
// 1024 threads * 8 elements; LDS Hillis-Steele over per-thread sums.
// ---------------------------------------------------------------------------
__global__ __launch_bounds__(1024) void scan_kernel(const float* __restrict__ mov,
                                                    float* __restrict__ cum) {
    __shared__ float lds[1024];
    const int t    = threadIdx.x;
    const int base = t * 8;

    float loc[8];
    float run = 0.0f;
#pragma unroll
    for (int i = 0; i < 8; ++i) {
        float v = (base + i < NPAIRS) ? mov[base + i] : 0.0f;
        run += v;
        loc[i] = run;
    }
    lds[t] = run;
    __syncthreads();

    for (int off = 1; off < 1024; off <<= 1) {
        float v = (t >= off) ? lds[t - off] : 0.0f;
        __syncthreads();
        lds[t] += v;
        __syncthreads();
    }
    const float excl = lds[t] - run;

#pragma unroll
    for (int i = 0; i < 8; ++i) {
        int idx = base + i;
        if (idx < NPAIRS) cum[idx + 1] = excl + loc[i];
    }
    if (t == 0) cum[0] = 0.0f;
}

// ---------------------------------------------------------------------------
// Kernel 3: one block per output frame. Thread 0: target, searchsorted-right,
// fractional index; broadcast via LDS; then coalesced lerp of the ORIGINAL
// (NaN-preserving) tensor frames.
// ---------------------------------------------------------------------------
__global__ __launch_bounds__(256) void resample_kernel(const float* __restrict__ in,
                                                       const float* __restrict__ mov,
                                                       const float* __restrict__ cum,
                                                       float* __restrict__ out) {
    __shared__ int   s_lo, s_hi;
    __shared__ float s_w;

    if (threadIdx.x == 0) {
        const float total  = cum[FRAMES - 1];
        const float target = (total / (float)(MAXF - 1)) * (float)blockIdx.x;

        // searchsorted(cum, target, side='right') over cum[0..8191]
        int lo = 0, hi = FRAMES;
        while (lo < hi) {
            int mid = (lo + hi) >> 1;
            if (cum[mid] <= target) lo = mid + 1; else hi = mid;
        }
        int j = lo;
        if (j < 1)          j = 1;
        if (j > FRAMES - 1) j = FRAMES - 1;

        const float seg  = mov[j - 1];
        const float frac = (target - cum[j - 1]) / (seg == 0.0f ? 1.0f : seg);
        float fidx = (float)(j - 1) + frac;
        if (fidx < 0.0f) fidx = 0.0f;
        const float fmax = (float)(FRAMES - 1);
        if (fidx > fmax) fidx = fmax;

        const float fl = floorf(fidx);
        s_lo = (int)fl;
        s_hi = (int)ceilf(fidx);
        s_w  = fidx - fl;
    }
    __syncthreads();

    const int   lo = s_lo, hi = s_hi;
    const float w  = s_w;
    const float* pl = in  + (size_t)lo * FRAME_ELEMS;
    const float* ph = in  + (size_t)hi * FRAME_ELEMS;
    float*       po = (float*)out + (size_t)blockIdx.x * FRAME_ELEMS;

    for (int e = threadIdx.x; e < FRAME_ELEMS; e += 256) {
        po[e] = pl[e] * (1.0f - w) + ph[e] * w;
    }
}

extern "C" void kernel_launch(void* const* d_in, const int* in_sizes, int n_in,
                              void* d_out, int out_size, void* d_ws, size_t ws_size,
                              hipStream_t stream) {
    const float* in  = (const float*)d_in[0];
    float*       out = (float*)d_out;
    float*       mov = (float*)d_ws;        // 8192 floats (8191 used)
    float*       cum = mov + FRAMES;        // 8192 floats; total 64 KB of d_ws

    // 8192 pair-slots = 128 blocks * 4 waves * 16 pairs (1 tail slot masked)
    movements_kernel<<<128, 128, 0, stream>>>(in, mov);
    scan_kernel<<<1, 1024, 0, stream>>>(mov, cum);
    resample_kernel<<<MAXF, 256, 0, stream>>>(in, mov, cum, out);
}